// OneDAffineTransformation_72885595013742
// MI455X (gfx1250) — compile-verified
//
#include <hip/hip_runtime.h>

typedef float v4f __attribute__((ext_vector_type(4)));
typedef int   v4i __attribute__((ext_vector_type(4)));

#define EMBED_DIM 4096
#define NTHREADS  256
#define V4_COUNT  (EMBED_DIM / 4)        // 1024 float4 per row
#define V4_PER_T  (V4_COUNT / NTHREADS)  // 4 float4 per thread

__global__ __launch_bounds__(NTHREADS)
void rank1_affine_kernel(const float* __restrict__ base,
                         const float* __restrict__ v,
                         const float* __restrict__ beta_p,
                         float* __restrict__ out)
{
    __shared__ v4f   lds_v[V4_COUNT];          // 16 KB staged copy of v
    __shared__ float red_rv[NTHREADS / 32];
    __shared__ float red_vv[NTHREADS / 32];

    const int    t       = threadIdx.x;
    const size_t row_off = (size_t)blockIdx.x * EMBED_DIM;

    const v4f* __restrict__ g_b4 = (const v4f*)(base + row_off);
    const v4f* __restrict__ g_v4 = (const v4f*)v;
    v4f*       __restrict__ g_o4 = (v4f*)(out + row_off);

    // ---- stage v into LDS via gfx1250 async global->LDS (ASYNCcnt path) ----
#if __has_builtin(__builtin_amdgcn_global_load_async_to_lds_b128)
#define ASYNC_V 1
#pragma unroll
    for (int i = 0; i < V4_PER_T; ++i) {
        const int j = t + NTHREADS * i;
        __builtin_amdgcn_global_load_async_to_lds_b128(
            (__attribute__((address_space(1))) v4i*)(g_v4 + j),
            (__attribute__((address_space(3))) v4i*)(&lds_v[j]),
            0, 0);
    }
#else
#pragma unroll
    for (int i = 0; i < V4_PER_T; ++i) {
        const int j = t + NTHREADS * i;
        lds_v[j] = g_v4[j];
    }
#endif

    // ---- pull this block's row of base into registers (overlaps async v) ----
    v4f b[V4_PER_T];
#pragma unroll
    for (int i = 0; i < V4_PER_T; ++i)
        b[i] = __builtin_nontemporal_load(g_b4 + t + NTHREADS * i);

    const float beta = beta_p[0];

#ifdef ASYNC_V
    asm volatile("s_wait_asynccnt 0" ::: "memory");
#endif
    __syncthreads();   // all waves' LDS writes of v are complete + visible

    // v slices into registers: reused for dot product AND final FMA
    v4f vv[V4_PER_T];
#pragma unroll
    for (int i = 0; i < V4_PER_T; ++i) vv[i] = lds_v[t + NTHREADS * i];

    float rv = 0.0f, vsq = 0.0f;
#pragma unroll
    for (int i = 0; i < V4_PER_T; ++i) {
#pragma unroll
        for (int c = 0; c < 4; ++c) {
            rv  = fmaf(b[i][c],  vv[i][c], rv);
            vsq = fmaf(vv[i][c], vv[i][c], vsq);
        }
    }

    // ---- wave32 butterfly reduction ----
#pragma unroll
    for (int m = 16; m >= 1; m >>= 1) {
        rv  += __shfl_xor(rv,  m, 32);
        vsq += __shfl_xor(vsq, m, 32);
    }
    const int wid  = t >> 5;
    const int lane = t & 31;
    if (lane == 0) { red_rv[wid] = rv; red_vv[wid] = vsq; }
    __syncthreads();

    float trv = 0.0f, tvv = 0.0f;
#pragma unroll
    for (int w = 0; w < NTHREADS / 32; ++w) { trv += red_rv[w]; tvv += red_vv[w]; }

    const float scale = beta - trv / tvv;

    // ---- out = base + scale * v, streaming stores ----
#pragma unroll
    for (int i = 0; i < V4_PER_T; ++i) {
        v4f o;
#pragma unroll
        for (int c = 0; c < 4; ++c) o[c] = fmaf(scale, vv[i][c], b[i][c]);
        __builtin_nontemporal_store(o, g_o4 + t + NTHREADS * i);
    }
}

extern "C" void kernel_launch(void* const* d_in, const int* in_sizes, int n_in,
                              void* d_out, int out_size, void* d_ws, size_t ws_size,
                              hipStream_t stream)
{
    (void)n_in; (void)out_size; (void)d_ws; (void)ws_size;
    const float* base = (const float*)d_in[0];
    const float* v    = (const float*)d_in[1];
    const float* beta = (const float*)d_in[2];
    float*       out  = (float*)d_out;

    const int rows = in_sizes[0] / EMBED_DIM;   // BATCH * SEQ = 8192
    rank1_affine_kernel<<<rows, NTHREADS, 0, stream>>>(base, v, beta, out);
}